// MultiHeadSelfAttention_1_44272522887186
// MI455X (gfx1250) — compile-verified
//
#include <hip/hip_runtime.h>
#include <hip/hip_bf16.h>

typedef float v2f __attribute__((ext_vector_type(2)));
typedef float v8f __attribute__((ext_vector_type(8)));

// Problem constants: B=32, C=5, H=50, S=20, D=400, NH=20 -> d_k=20
// BCH = 8000, M = BCH*S = 160000, context = 160000 floats, q_a = 64e6 floats.

// ---------------------------------------------------------------------------
// Fold lin_w into the projection weights:
// Afold[f=(s*400+e)][p*20+t] = sum_m lin_w[s*20+m] * W_p[(m*20+t)*400 + e]
// so that  {Qr,Kr,Vr}[row,t] = sum_f Qflat[row,f]*Afold[f][..t] + cvec[..t]
// ---------------------------------------------------------------------------
__global__ void fold_kernel(const float* __restrict__ Wq, const float* __restrict__ Wk,
                            const float* __restrict__ Wv, const float* __restrict__ lin_w,
                            float* __restrict__ Afold) {
    int gid = blockIdx.x * 256 + threadIdx.x;
    if (gid < 3 * 160000) {
        int p = gid / 160000;
        int rem = gid % 160000;
        int f = rem / 20, t = rem % 20;
        int s = f / 400, e = f % 400;
        const float* W = (p == 0) ? Wq : (p == 1) ? Wk : Wv;
        float a = 0.f;
#pragma unroll
        for (int m = 0; m < 20; ++m)
            a += lin_w[s * 20 + m] * W[(m * 20 + t) * 400 + e];
        Afold[f * 64 + p * 20 + t] = a;
    } else if (gid < 3 * 160000 + 32000) {
        int z = gid - 3 * 160000;          // zero pad columns 60..63
        Afold[(z >> 2) * 64 + 60 + (z & 3)] = 0.f;
    }
}

__global__ void cbias_kernel(const float* __restrict__ bq, const float* __restrict__ bk,
                             const float* __restrict__ bv, const float* __restrict__ lin_w,
                             const float* __restrict__ lin_b, float* __restrict__ cvec) {
    int j = threadIdx.x;
    if (j < 60) {
        int p = j / 20, t = j % 20;
        const float* bb = (p == 0) ? bq : (p == 1) ? bk : bv;
        float acc = lin_b[0];
        for (int m = 0; m < 20; ++m) {
            float sm = 0.f;
            for (int s = 0; s < 20; ++s) sm += lin_w[s * 20 + m];
            acc += sm * bb[m * 20 + t];
        }
        cvec[j] = acc;
    } else if (j < 64) {
        cvec[j] = 0.f;
    }
}

// ---------------------------------------------------------------------------
// q_a = Q * Wq^T + bq   (M=160000, N=400, K=400)  via V_WMMA_F32_16X16X4_F32
// Block: 256 threads = 8 waves. Tile: 128 rows x 80 cols; wave w owns rows
// [w*16, w*16+16) and all 80 cols (5 accumulators). K staged 16 at a time.
// Grid is linearized with the 5 N-tiles innermost so re-reads of a Q row
// panel land in L2 (Q = 256 MB > streams HBM only once).
// B is staged pair-interleaved (Bs[k/2][n][2]) so each lane's (K,K+1) pair
// is one aligned ds_load_b64 into an adjacent VGPR pair (no v_mov packing).
// ---------------------------------------------------------------------------
__global__ __launch_bounds__(256)
void gemm_qa_kernel(const float* __restrict__ Q, const float* __restrict__ Wq,
                    const float* __restrict__ bq, float* __restrict__ qa) {
    __shared__ float As[128 * 18];    // row pad 16->18 (even: keeps b64 loads aligned)
    __shared__ float Bs[8 * 80 * 2];  // [kpair][n][2]
    const int tid  = threadIdx.x;
    const int wave = tid >> 5;
    const int lane = tid & 31;
    const int half = lane >> 4;       // 0: K+0/1, 1: K+2/3 (ISA 16x16x4 A/B layout)
    const int lm   = lane & 15;
    const int bid  = blockIdx.x;      // 0..6249, N-tiles innermost
    const int row0 = (bid / 5) * 128;
    const int n0   = (bid % 5) * 80;

    v8f zero = {0, 0, 0, 0, 0, 0, 0, 0};
    v8f acc[5];
#pragma unroll
    for (int i = 0; i < 5; ++i) acc[i] = zero;

    for (int kk = 0; kk < 400; kk += 16) {
        for (int idx = tid; idx < 128 * 16; idx += 256) {
            int r = idx >> 4, c = idx & 15;
            As[r * 18 + c] = Q[(size_t)(row0 + r) * 400 + kk + c];
        }
        // Bs[k/2][n][k&1] = Wq^T[kk+k, n0+n] = Wq[(n0+n)*400 + kk+k]
        for (int idx = tid; idx < 16 * 80; idx += 256) {
            int k = idx & 15, n = idx >> 4;
            Bs[((k >> 1) * 80 + n) * 2 + (k & 1)] = Wq[(n0 + n) * 400 + kk + k];
        }
        __syncthreads();
#pragma unroll
        for (int kb = 0; kb < 16; kb += 4) {
            int acol = kb + half * 2;
            v2f a = *(const v2f*)&As[(wave * 16 + lm) * 18 + acol];
            int P = (kb >> 1) + half;            // K-pair index for this half-wave
#pragma unroll
            for (int nt = 0; nt < 5; ++nt) {
                v2f b = *(const v2f*)&Bs[(P * 80 + nt * 16 + lm) * 2];
                acc[nt] = __builtin_amdgcn_wmma_f32_16x16x4_f32(
                    false, a, false, b, (short)0, acc[nt], false, false);
            }
        }
        __syncthreads();
    }
    // C/D layout: VGPR r, lanes0-15 -> M=r, lanes16-31 -> M=r+8; N = lane%16
#pragma unroll
    for (int nt = 0; nt < 5; ++nt) {
#pragma unroll
        for (int r = 0; r < 8; ++r) {
            int row = row0 + wave * 16 + r + half * 8;
            int col = n0 + nt * 16 + lm;
            qa[(size_t)row * 400 + col] = acc[nt][r] + bq[col];
        }
    }
}

// ---------------------------------------------------------------------------
// [Qr|Kr|Vr] = Qflat * Afold + cvec   (M=8000, N=64, K=8000)
// Block: 128 threads = 4 waves; tile 64 rows x 64 cols.
// ---------------------------------------------------------------------------
__global__ __launch_bounds__(128)
void gemm_r_kernel(const float* __restrict__ Q, const float* __restrict__ Afold,
                   const float* __restrict__ cvec, float* __restrict__ Qr,
                   float* __restrict__ Kr, float* __restrict__ Vr) {
    __shared__ float As[64 * 18];
    __shared__ float Bs[8 * 64 * 2];  // [kpair][n][2]
    const int tid  = threadIdx.x;
    const int wave = tid >> 5;
    const int lane = tid & 31;
    const int half = lane >> 4;
    const int lm   = lane & 15;
    const int row0 = blockIdx.x * 64;

    v8f zero = {0, 0, 0, 0, 0, 0, 0, 0};
    v8f acc[4];
#pragma unroll
    for (int i = 0; i < 4; ++i) acc[i] = zero;

    for (int kk = 0; kk < 8000; kk += 16) {
        for (int idx = tid; idx < 64 * 16; idx += 128) {
            int r = idx >> 4, c = idx & 15;
            As[r * 18 + c] = Q[(size_t)(row0 + r) * 8000 + kk + c];
        }
        for (int idx = tid; idx < 16 * 64; idx += 128) {
            int k = idx >> 6, n = idx & 63;
            Bs[((k >> 1) * 64 + n) * 2 + (k & 1)] = Afold[(kk + k) * 64 + n];
        }
        __syncthreads();
#pragma unroll
        for (int kb = 0; kb < 16; kb += 4) {
            int acol = kb + half * 2;
            v2f a = *(const v2f*)&As[(wave * 16 + lm) * 18 + acol];
            int P = (kb >> 1) + half;
#pragma unroll
            for (int nt = 0; nt < 4; ++nt) {
                v2f b = *(const v2f*)&Bs[(P * 64 + nt * 16 + lm) * 2];
                acc[nt] = __builtin_amdgcn_wmma_f32_16x16x4_f32(
                    false, a, false, b, (short)0, acc[nt], false, false);
            }
        }
        __syncthreads();
    }
#pragma unroll
    for (int nt = 0; nt < 4; ++nt) {
#pragma unroll
        for (int r = 0; r < 8; ++r) {
            int row3 = row0 + wave * 16 + r + half * 8;
            int j    = nt * 16 + lm;
            float v  = acc[nt][r] + cvec[j];
            if (j < 20)      Qr[row3 * 20 + j]      = v;
            else if (j < 40) Kr[row3 * 20 + j - 20] = v;
            else if (j < 60) Vr[row3 * 20 + j - 40] = v;
        }
    }
}

// ---------------------------------------------------------------------------
// Attention per (b,c): scores = exp((Qr Kr^T + Kr cddr^T)/sqrt(20)),
// attn = scores / (rowsum + 1e-8), context = attn * Vr.
// ---------------------------------------------------------------------------
__global__ __launch_bounds__(256)
void attn_kernel(const float* __restrict__ cdd, const float* __restrict__ lin1_w,
                 const float* __restrict__ lin1_b, const float* __restrict__ Qr,
                 const float* __restrict__ Kr, const float* __restrict__ Vr,
                 float* __restrict__ ctx) {
    __shared__ float sQ[1000], sK[1000], sV[1000], sC[1000], sS[2500], sInv[50];
    const int tid = threadIdx.x;
    const int bc  = blockIdx.x;               // (b*C + c), 0..159
    const float scale = 0.22360679774997896f; // 1/sqrt(20)

    for (int idx = tid; idx < 1000; idx += 256) {
        sQ[idx] = Qr[bc * 1000 + idx];
        sK[idx] = Kr[bc * 1000 + idx];
        sV[idx] = Vr[bc * 1000 + idx];
        int h = idx / 20, t = idx % 20;
        float a = lin1_b[0];
#pragma unroll
        for (int n = 0; n < 20; ++n)
            a += cdd[(size_t)(bc * 50 + h) * 400 + n * 20 + t] * lin1_w[n];
        sC[idx] = a;
    }
    __syncthreads();
    for (int idx = tid; idx < 2500; idx += 256) {
        int i = idx / 50, j = idx % 50;
        float s1 = 0.f, s2 = 0.f;
#pragma unroll
        for (int t = 0; t < 20; ++t) {
            s1 += sQ[i * 20 + t] * sK[j * 20 + t];
            s2 += sK[i * 20 + t] * sC[j * 20 + t];
        }
        sS[idx] = expf((s1 + s2) * scale);    // faithful: no max-subtraction
    }
    __syncthreads();
    if (tid < 50) {
        float s = 0.f;
        for (int j = 0; j < 50; ++j) s += sS[tid * 50 + j];
        sInv[tid] = 1.f / (s + 1e-8f);
    }
    __syncthreads();
    for (int idx = tid; idx < 1000; idx += 256) {
        int i = idx / 20, t = idx % 20;
        float a = 0.f;
        for (int j = 0; j < 50; ++j) a += sS[i * 50 + j] * sV[j * 20 + t];
        ctx[bc * 1000 + idx] = a * sInv[i];
    }
}

// ---------------------------------------------------------------------------
extern "C" void kernel_launch(void* const* d_in, const int* in_sizes, int n_in,
                              void* d_out, int out_size, void* d_ws, size_t ws_size,
                              hipStream_t stream) {
    (void)in_sizes; (void)n_in; (void)out_size; (void)ws_size;
    const float* Q      = (const float*)d_in[0];
    const float* cdd    = (const float*)d_in[1];
    const float* Wq     = (const float*)d_in[2];
    const float* bq     = (const float*)d_in[3];
    const float* Wk     = (const float*)d_in[4];
    const float* bk     = (const float*)d_in[5];
    const float* Wv     = (const float*)d_in[6];
    const float* bv     = (const float*)d_in[7];
    const float* lin_w  = (const float*)d_in[8];
    const float* lin_b  = (const float*)d_in[9];
    const float* lin1_w = (const float*)d_in[10];
    const float* lin1_b = (const float*)d_in[11];

    float* out = (float*)d_out;
    float* ctx = out;             // context: 160000 floats
    float* qa  = out + 160000;    // q_a:     64e6 floats

    // workspace layout (floats): Afold 8000*64, cvec 64, Qr/Kr/Vr 160000 each (~3.97 MB)
    float* ws    = (float*)d_ws;
    float* Afold = ws;
    float* cvec  = ws + 512000;
    float* QrW   = cvec + 64;
    float* KrW   = QrW + 160000;
    float* VrW   = KrW + 160000;

    fold_kernel<<<2000, 256, 0, stream>>>(Wq, Wk, Wv, lin_w, Afold);
    cbias_kernel<<<1, 64, 0, stream>>>(bq, bk, bv, lin_w, lin_b, cvec);

    gemm_qa_kernel<<<6250, 256, 0, stream>>>(Q, Wq, bq, qa);   // 1250 M-blocks x 5 N-tiles (N inner)
    gemm_r_kernel<<<125, 128, 0, stream>>>(Q, Afold, cvec, QrW, KrW, VrW);
    attn_kernel<<<160, 256, 0, stream>>>(cdd, lin1_w, lin1_b, QrW, KrW, VrW, ctx);
}